// OnionPeelHead_90117003804897
// MI455X (gfx1250) — compile-verified
//
#include <hip/hip_runtime.h>
#include <hip/hip_bf16.h>
#include <math.h>

typedef __attribute__((ext_vector_type(2))) float v2f;
typedef __attribute__((ext_vector_type(8))) float v8f;

#define Bb 16
#define Tt 4096
#define Dm 1024
#define Cc 1000
#define Kk 4
#define TEMP_INV (1.0f / 0.07f)

// workspace layout (in floats)
#define U_OFF 0          // 4*1024
#define G_OFF 4096       // 16
#define W_OFF 4112       // 4*1000
#define S_OFF 8112       // 4*16
#define C_OFF 8192       // 4*16*4096

// ---------------- k0: u_k = (v*sigmoid(m))/||.||, G = U U^T ----------------
__global__ void k0_prep(const float* __restrict__ v, const float* __restrict__ m_logits,
                        float* __restrict__ Uws, float* __restrict__ Gws) {
    __shared__ float red[256];
    int tid = threadIdx.x;
    for (int k = 0; k < Kk; ++k) {
        float vk[4];
        float part = 0.f;
        #pragma unroll
        for (int i = 0; i < 4; ++i) {
            int d = tid * 4 + i;
            float m = m_logits[k * Dm + d];
            float sig = 1.0f / (1.0f + __expf(-m));
            vk[i] = v[k * Dm + d] * sig;
            part += vk[i] * vk[i];
        }
        red[tid] = part; __syncthreads();
        for (int s = 128; s > 0; s >>= 1) { if (tid < s) red[tid] += red[tid + s]; __syncthreads(); }
        float inv = 1.0f / (sqrtf(red[0]) + 1e-6f);
        __syncthreads();
        #pragma unroll
        for (int i = 0; i < 4; ++i) Uws[k * Dm + tid * 4 + i] = vk[i] * inv;
        __syncthreads();
    }
    // Gram matrix G[j][k] = u_j . u_k
    for (int j = 0; j < Kk; ++j) {
        for (int k = 0; k < Kk; ++k) {
            float part = 0.f;
            #pragma unroll
            for (int i = 0; i < 4; ++i) {
                int d = tid * 4 + i;
                part += Uws[j * Dm + d] * Uws[k * Dm + d];
            }
            red[tid] = part; __syncthreads();
            for (int s = 128; s > 0; s >>= 1) { if (tid < s) red[tid] += red[tid + s]; __syncthreads(); }
            if (tid == 0) Gws[j * 4 + k] = red[0];
            __syncthreads();
        }
    }
}

// ---- k1: single streaming pass C = E @ U^T via V_WMMA_F32_16X16X4_F32, ----
// ---- then per-token peel recurrence c_k = raw_k - sum_{j<k} b_j c_j G_jk ---
__global__ void k1_coeff(const float* __restrict__ E, const float* __restrict__ Uws,
                         const float* __restrict__ Gws, const float* __restrict__ beta,
                         float* __restrict__ Cp) {
    // 5 rows: u_0..u_3 plus an all-zero row for B-matrix columns 4..15
    __shared__ float ldsU[(Kk + 1) * Dm];   // 20 KB
    __shared__ float ldsC[8 * 256];         // 8 KB: one 16x16 f32 tile per wave
    int tid = threadIdx.x;
    for (int i = tid; i < (Kk + 1) * Dm; i += 256)
        ldsU[i] = (i < Kk * Dm) ? Uws[i] : 0.0f;
    __syncthreads();

    int wave = tid >> 5;
    int lane = tid & 31;
    int n = lane & 15;       // row (A) / col (B,C)
    int h = lane >> 4;       // lane half selects K sub-pair
    int tile = blockIdx.x * 8 + wave;          // 4096 tiles of 16 rows
    const float* Erow = E + ((long)tile * 16 + n) * Dm;
    // columns >= 4 read the zero row: no per-iteration VALU masking needed
    const float* up = &ldsU[(n < Kk ? n : Kk) * Dm];

    v8f acc = {};
    for (int d = 0; d < Dm; d += 4) {
        // A (16x4 f32): lane(n,h) vgpr r holds E[row n][d + 2h + r]
        // E is streamed once (256 MB > 192 MB L2): non-temporal load
        v2f a = __builtin_nontemporal_load((const v2f*)(Erow + d + 2 * h));
        // B (4x16 f32): lane(n,h) vgpr r holds U[col n][d + 2h + r]
        v2f bm;
        bm.x = up[d + 2 * h];
        bm.y = up[d + 2 * h + 1];
        acc = __builtin_amdgcn_wmma_f32_16x16x4_f32(false, a, false, bm,
                                                    (short)0, acc, false, false);
    }
    // spill C tile (lane(n,h) vgpr r -> element M=r+8h, N=n)
    float* myTile = &ldsC[wave * 256];
    #pragma unroll
    for (int r = 0; r < 8; ++r) myTile[(r + 8 * h) * 16 + n] = acc[r];
    __syncthreads();

    if (lane < 16) {
        int row = lane;
        int b  = tile >> 8;                    // 256 tiles per batch row
        int t  = ((tile & 255) << 4) + row;
        float c0 = myTile[row * 16 + 0];
        float c1 = myTile[row * 16 + 1];
        float c2 = myTile[row * 16 + 2];
        float c3 = myTile[row * 16 + 3];
        float be0 = beta[0], be1 = beta[1], be2 = beta[2];
        c1 -= be0 * c0 * Gws[0 * 4 + 1];
        c2 -= be0 * c0 * Gws[0 * 4 + 2] + be1 * c1 * Gws[1 * 4 + 2];
        c3 -= be0 * c0 * Gws[0 * 4 + 3] + be1 * c1 * Gws[1 * 4 + 3] + be2 * c2 * Gws[2 * 4 + 3];
        long bt = (long)b * Tt + t;
        // Cp (1 MB) is re-read by k2 soon after (k3 runs in between): keep regular-temporal so it
        // stays resident in the 192 MB L2
        Cp[0L * Bb * Tt + bt] = c0;
        Cp[1L * Bb * Tt + bt] = c1;
        Cp[2L * Bb * Tt + bt] = c2;
        Cp[3L * Bb * Tt + bt] = c3;
    }
}

// ------ k2: per (b,k) softmax-weighted sum + top-8 sum over T=4096 ---------
__global__ void k2_reduce(const float* __restrict__ Cp, float* __restrict__ Sws) {
    __shared__ float cache[Tt];     // 16 KB slice of coefficients
    __shared__ float red[256];
    __shared__ int   redi[256];
    int tid = threadIdx.x;
    int k = blockIdx.x >> 4;
    int b = blockIdx.x & 15;
    const float* src = Cp + ((long)k * Bb + b) * Tt;

    float lmax = -__builtin_inff();
    for (int i = tid; i < Tt; i += 256) { float c = src[i]; cache[i] = c; lmax = fmaxf(lmax, c); }
    red[tid] = lmax; __syncthreads();
    for (int s = 128; s > 0; s >>= 1) { if (tid < s) red[tid] = fmaxf(red[tid], red[tid + s]); __syncthreads(); }
    float cmax = red[0]; __syncthreads();

    float se = 0.f, sce = 0.f;
    for (int i = tid; i < Tt; i += 256) {
        float c = cache[i];
        float e = __expf((c - cmax) * TEMP_INV);
        se += e; sce += c * e;
    }
    red[tid] = se; __syncthreads();
    for (int s = 128; s > 0; s >>= 1) { if (tid < s) red[tid] += red[tid + s]; __syncthreads(); }
    float sumexp = red[0]; __syncthreads();
    red[tid] = sce; __syncthreads();
    for (int s = 128; s > 0; s >>= 1) { if (tid < s) red[tid] += red[tid + s]; __syncthreads(); }
    float sumce = red[0]; __syncthreads();
    float soft = sumce / sumexp;

    float topsum = 0.f;
    for (int pass = 0; pass < 8; ++pass) {
        float bm = -__builtin_inff(); int bi = 0;
        for (int i = tid; i < Tt; i += 256) { float c = cache[i]; if (c > bm) { bm = c; bi = i; } }
        red[tid] = bm; redi[tid] = bi; __syncthreads();
        for (int s = 128; s > 0; s >>= 1) {
            if (tid < s && red[tid + s] > red[tid]) { red[tid] = red[tid + s]; redi[tid] = redi[tid + s]; }
            __syncthreads();
        }
        if (tid == 0) { topsum += red[0]; cache[redi[0]] = -__builtin_inff(); }
        __syncthreads();
    }
    if (tid == 0) Sws[k * Bb + b] = 0.5f * (topsum + soft);
}

// --------- k3: W[k][c] = cls_W[k][c] . u_k  (one wave per output row) ------
__global__ void k3_clsw(const float* __restrict__ clsW, const float* __restrict__ Uws,
                        float* __restrict__ Wws) {
    int gid = blockIdx.x * blockDim.x + threadIdx.x;
    int wave = gid >> 5;
    int lane = gid & 31;
    if (wave >= Kk * Cc) return;
    int k = wave / Cc;
    const float* wr = clsW + (long)wave * Dm;
    const float* u  = Uws + k * Dm;
    float acc = 0.f;
    // cls_W (16 MB) is streamed once: non-temporal
    for (int j = lane; j < Dm; j += 32)
        acc = fmaf(__builtin_nontemporal_load(wr + j), u[j], acc);
    for (int off = 16; off > 0; off >>= 1) acc += __shfl_down(acc, off, 32);
    if (lane == 0) Wws[wave] = acc;
}

// --------- k4: out[b,c] = sum_k alpha_k (s_k[b] W_k[c] + cls_b[k,c]) -------
__global__ void k4_final(const float* __restrict__ Wws, const float* __restrict__ Sws,
                         const float* __restrict__ cls_b, const float* __restrict__ alpha,
                         float* __restrict__ out) {
    int idx = blockIdx.x * blockDim.x + threadIdx.x;
    if (idx >= Bb * Cc) return;
    int b = idx / Cc, c = idx % Cc;
    float r = 0.f;
    #pragma unroll
    for (int k = 0; k < Kk; ++k)
        r += alpha[k] * (Sws[k * Bb + b] * Wws[k * Cc + c] + cls_b[k * Cc + c]);
    out[idx] = r;
}

extern "C" void kernel_launch(void* const* d_in, const int* in_sizes, int n_in,
                              void* d_out, int out_size, void* d_ws, size_t ws_size,
                              hipStream_t stream) {
    const float* E        = (const float*)d_in[0];
    const float* v        = (const float*)d_in[1];
    const float* m_logits = (const float*)d_in[2];
    const float* cls_W    = (const float*)d_in[3];
    const float* cls_b    = (const float*)d_in[4];
    const float* beta     = (const float*)d_in[5];
    const float* alpha    = (const float*)d_in[6];
    float* out = (float*)d_out;
    float* ws  = (float*)d_ws;

    float* Uws = ws + U_OFF;
    float* Gws = ws + G_OFF;
    float* Wws = ws + W_OFF;
    float* Sws = ws + S_OFF;
    float* Cp  = ws + C_OFF;

    k0_prep<<<1, 256, 0, stream>>>(v, m_logits, Uws, Gws);
    k1_coeff<<<(Bb * Tt / 16) / 8, 256, 0, stream>>>(E, Uws, Gws, beta, Cp);
    k3_clsw<<<(Kk * Cc * 32) / 256, 256, 0, stream>>>(cls_W, Uws, Wws);
    k2_reduce<<<Kk * Bb, 256, 0, stream>>>(Cp, Sws);
    k4_final<<<(Bb * Cc + 255) / 256, 256, 0, stream>>>(Wws, Sws, cls_b, alpha, out);
}